// Sample_loss_5669356832499
// MI455X (gfx1250) — compile-verified
//
#include <hip/hip_runtime.h>
#include <hip/hip_bf16.h>

typedef __attribute__((ext_vector_type(2))) float v2f;
typedef __attribute__((ext_vector_type(8))) float v8f;

// ---------------------------------------------------------------------------
// Kernel 1: one wave32 per row. p = prod_{j < len} (1 - x[b, y[b,j]]),
// write ws[b] = 1 - p.  Rows in [n, nRound) are zero-filled so the WMMA
// reduction can consume a multiple of 64 values.
// ---------------------------------------------------------------------------
__global__ void row_prod_kernel(const float* __restrict__ x,
                                const int*   __restrict__ y,
                                const int*   __restrict__ lengths,
                                float* __restrict__ ws,
                                int n, int nRound, int C, int L) {
    const int wave = (int)((blockIdx.x * blockDim.x + threadIdx.x) >> 5);
    const int lane = (int)(threadIdx.x & 31);
    if (wave >= nRound) return;
    if (wave >= n) {                       // padding rows -> 0 contribution
        if (lane == 0) ws[wave] = 0.0f;
        return;
    }

    const int len = lengths[wave];
    const int*   yr = y + (long long)wave * L;
    const float* xr = x + (long long)wave * C;

    float p = 1.0f;
    for (int j = lane; j < len; j += 32) {
        const int idx = yr[j];
        p *= (1.0f - xr[idx]);
    }
    // wave32 tree product
    #pragma unroll
    for (int off = 16; off > 0; off >>= 1)
        p *= __shfl_xor(p, off, 32);

    if (lane == 0) ws[wave] = 1.0f - p;
}

// ---------------------------------------------------------------------------
// Kernel 2: single wave32. Sum nRound floats (multiple of 64) with
// V_WMMA_F32_16X16X4_F32:  A = ones(16x4), B = 4x16 chunk of values,
// C accumulates column sums. All 16 rows of D are identical, and lanes
// 16..31 of c[0] duplicate lanes 0..15, so a full 32-lane add of c[0]
// yields exactly 2x the total; *0.5f is exact.
// ---------------------------------------------------------------------------
__global__ void wmma_sum_kernel(const float* __restrict__ ws,
                                float* __restrict__ out,
                                int nRound) {
    const int lane = (int)threadIdx.x;   // blockDim.x == 32, EXEC all ones

    v2f a; a.x = 1.0f; a.y = 1.0f;       // A-matrix: all ones
    v8f c = {};                          // C/D accumulator, starts at 0

    for (int i = 0; i < nRound; i += 64) {
        const int base = i + lane * 2;
        v2f b;
        b.x = ws[base];
        b.y = ws[base + 1];
        // 8 args: (neg_a, A, neg_b, B, c_mod, C, reuse_a, reuse_b)
        c = __builtin_amdgcn_wmma_f32_16x16x4_f32(
                false, a, false, b, (short)0, c, false, false);
    }

    float s = c[0];
    #pragma unroll
    for (int off = 16; off > 0; off >>= 1)
        s += __shfl_xor(s, off, 32);

    if (lane == 0) out[0] = 0.5f * s;
}

// ---------------------------------------------------------------------------
// Launch
// ---------------------------------------------------------------------------
extern "C" void kernel_launch(void* const* d_in, const int* in_sizes, int n_in,
                              void* d_out, int out_size, void* d_ws, size_t ws_size,
                              hipStream_t stream) {
    const float* x       = (const float*)d_in[0];   // (B, C) f32
    const int*   y       = (const int*)d_in[1];     // (B, L) int
    const int*   lengths = (const int*)d_in[2];     // (B,)   int

    const int B = in_sizes[2];
    const int C = in_sizes[0] / B;
    const int L = in_sizes[1] / B;
    const int n = B / 8;                            // rows actually used
    const int nRound = (n + 63) & ~63;              // multiple of 64 for WMMA

    float* ws  = (float*)d_ws;
    float* out = (float*)d_out;

    // Kernel 1: one wave32 per row, 8 waves per 256-thread block.
    const int wavesNeeded = nRound;
    const int blocks = (wavesNeeded * 32 + 255) / 256;
    row_prod_kernel<<<blocks, 256, 0, stream>>>(x, y, lengths, ws, n, nRound, C, L);

    // Kernel 2: single-wave WMMA reduction of ws[0..nRound) -> out[0].
    wmma_sum_kernel<<<1, 32, 0, stream>>>(ws, out, nRound);
}